// Model_51556787421441
// MI455X (gfx1250) — compile-verified
//
#include <hip/hip_runtime.h>

typedef float v2f __attribute__((ext_vector_type(2)));
typedef float v8f __attribute__((ext_vector_type(8)));
typedef int v4i_vs __attribute__((vector_size(16)));  // matches builtin's pointee type

#define B_SZ   2048
#define N_INST 16
#define N_FEAT 1024
#define KDIM   1024
#define N_HID  256

// Block tile
#define BM 128
#define BN 128
#define BK 32
#define LDA_S 36    // As row stride (floats): rows 16B-aligned (36*4=144), b64-frag friendly

#if defined(__has_builtin)
#if __has_builtin(__builtin_amdgcn_global_load_async_to_lds_b128)
#define HAVE_ASYNC_LDS 1
#endif
#if __has_builtin(__builtin_amdgcn_s_wait_asynccnt)
#define HAVE_WAIT_ASYNC_BUILTIN 1
#endif
#endif

__device__ __forceinline__ void wait_async0() {
#ifdef HAVE_WAIT_ASYNC_BUILTIN
  __builtin_amdgcn_s_wait_asynccnt(0);
#else
  asm volatile("s_wait_asynccnt 0" ::: "memory");
#endif
}

#ifdef HAVE_ASYNC_LDS
__device__ __forceinline__ void async_g2l_b128(const float* g, float* l) {
  __builtin_amdgcn_global_load_async_to_lds_b128(
      (__attribute__((address_space(1))) v4i_vs*)g,
      (__attribute__((address_space(3))) v4i_vs*)l, 0, 0);
}
#endif

// ---------------------------------------------------------------------------
// Column-normalize A over the feature axis: nA[i][f][k] = A[i][f][k]/||A[i][:,k]||
// ---------------------------------------------------------------------------
__global__ __launch_bounds__(256) void norm_cols_kernel(const float* __restrict__ A,
                                                        float* __restrict__ nA) {
  const int i = blockIdx.y;
  const int k = blockIdx.x * 256 + threadIdx.x;
  const float* Ai = A + (size_t)i * N_FEAT * KDIM;
  float* nAi = nA + (size_t)i * N_FEAT * KDIM;

  float acc = 0.f;
#pragma unroll 4
  for (int f = 0; f < N_FEAT; ++f) {
    float v = Ai[(size_t)f * KDIM + k];
    acc += v * v;
  }
  const float inv = 1.0f / sqrtf(acc);
#pragma unroll 4
  for (int f = 0; f < N_FEAT; ++f) {
    nAi[(size_t)f * KDIM + k] = Ai[(size_t)f * KDIM + k] * inv;
  }
}

// ---------------------------------------------------------------------------
// Generic batched (per-instance) fp32 WMMA GEMM: C[i] = A[i] * opB(B[i])
//   A: M x Kd row-major (contiguous along Kd), row stride lda, instance stride sAi
//   B (BT=0): Kd x N row-major, ldb = row stride
//   B (BT=1): logical Kd x N where B[k][n] = Bmem[n*ldb + k]
//   FUSE: C = acc + bias[col] (pre_relu); C2 = relu(C)
// Block: 256 threads = 8 waves; block tile 128x128x32; wave tile 32x64 (2x4 WMMAs).
// LDS layout: As m-major [m][k]; Bs n-major [n][k]  -> both WMMA fragments are
// contiguous b64 LDS reads (no VGPR repacking in the inner loop).
// ---------------------------------------------------------------------------
template <bool BT, bool FUSE>
__global__ __launch_bounds__(256) void gemm_f32_wmma_kernel(
    const float* __restrict__ Ab, long long sAi, int lda,
    const float* __restrict__ Bb, long long sBi, int ldb,
    float* __restrict__ Cb, long long sCi, int ldc,
    const float* __restrict__ biasb, float* __restrict__ C2b, int Kd) {
  constexpr int LDBS = BT ? 36 : 34;  // BT: 16B-aligned rows for async b128; !BT: fewer scatter conflicts
  __shared__ float As[BM * LDA_S];
  __shared__ float Bs[BN * LDBS];

  const int i = blockIdx.z;
  const float* Amat = Ab + (size_t)i * sAi;
  const float* Bmat = Bb + (size_t)i * sBi;
  float* C = Cb + (size_t)i * sCi;
  float* C2 = nullptr;
  const float* bias = nullptr;
  if (FUSE) {
    C2 = C2b + (size_t)i * sCi;
    bias = biasb + (size_t)i * N_FEAT;
  }

  const int m0 = blockIdx.y * BM;
  const int n0 = blockIdx.x * BN;

  const int tid = threadIdx.x;
  const int lane = tid & 31;
  const int wave = tid >> 5;
  const int waveM = wave & 3;  // 4 waves along M -> 32 rows each
  const int waveN = wave >> 2; // 2 waves along N -> 64 cols each
  const int hi = lane >> 4;    // lane half: K{0,1} vs K{2,3} (A/B), row half (C)
  const int lm = lane & 15;

  v8f acc[2][4];
#pragma unroll
  for (int sm = 0; sm < 2; ++sm)
#pragma unroll
    for (int sn = 0; sn < 4; ++sn) acc[sm][sn] = (v8f){};

  for (int kt = 0; kt < Kd; kt += BK) {
    // ---- Stage A tile: BM x BK, m-major (global contiguous along k) ----
    {
      const int r = tid >> 3;        // 0..31
      const int c4 = (tid & 7) * 4;  // 0..28
#pragma unroll
      for (int it = 0; it < 4; ++it) {
        const int row = r + it * 32;
        const float* gsrc = Amat + (size_t)(m0 + row) * lda + kt + c4;
        float* ldst = &As[row * LDA_S + c4];
#ifdef HAVE_ASYNC_LDS
        async_g2l_b128(gsrc, ldst);
#else
        *reinterpret_cast<float4*>(ldst) = *reinterpret_cast<const float4*>(gsrc);
#endif
      }
      if (kt + BK < Kd) {
        __builtin_prefetch(Amat + (size_t)(m0 + r) * lda + kt + BK + c4, 0, 0);
      }
    }
    // ---- Stage B tile: stored n-major Bs[n][k] ----
    if (BT) {
      // memory is n-major already: contiguous along k -> direct (async) copy
      const int n = tid >> 3;        // 0..31
      const int k4 = (tid & 7) * 4;  // 0..28
#pragma unroll
      for (int it = 0; it < 4; ++it) {
        const int col = n + it * 32;
        const float* gsrc = Bmat + (size_t)(n0 + col) * ldb + kt + k4;
        float* ldst = &Bs[col * LDBS + k4];
#ifdef HAVE_ASYNC_LDS
        async_g2l_b128(gsrc, ldst);
#else
        *reinterpret_cast<float4*>(ldst) = *reinterpret_cast<const float4*>(gsrc);
#endif
      }
    } else {
      // memory is k-major: coalesced float4 along n, scatter-transpose into Bs[n][k]
      const int nn = (tid & 31) * 4;  // 0..124
      const int kk = tid >> 5;        // 0..7
#pragma unroll
      for (int it = 0; it < 4; ++it) {
        const int k = kk + it * 8;
        const float4 v = *reinterpret_cast<const float4*>(
            Bmat + (size_t)(kt + k) * ldb + n0 + nn);
        Bs[(nn + 0) * LDBS + k] = v.x;
        Bs[(nn + 1) * LDBS + k] = v.y;
        Bs[(nn + 2) * LDBS + k] = v.z;
        Bs[(nn + 3) * LDBS + k] = v.w;
      }
    }
#ifdef HAVE_ASYNC_LDS
    wait_async0();
#endif
    __syncthreads();

    // ---- Compute: 8 WMMA k-steps of 4 over the BK=32 tile ----
#pragma unroll
    for (int k0 = 0; k0 < BK; k0 += 4) {
      const int ko = k0 + hi * 2;  // lanes 0-15 -> K{0,1}; lanes 16-31 -> K{2,3}
      v2f a[2], b[4];
#pragma unroll
      for (int sm = 0; sm < 2; ++sm) {
        const int row = waveM * 32 + sm * 16 + lm;
        a[sm] = *reinterpret_cast<const v2f*>(&As[row * LDA_S + ko]);
      }
#pragma unroll
      for (int sn = 0; sn < 4; ++sn) {
        const int col = waveN * 64 + sn * 16 + lm;
        b[sn] = *reinterpret_cast<const v2f*>(&Bs[col * LDBS + ko]);
      }
#pragma unroll
      for (int sm = 0; sm < 2; ++sm)
#pragma unroll
        for (int sn = 0; sn < 4; ++sn)
          acc[sm][sn] = __builtin_amdgcn_wmma_f32_16x16x4_f32(
              false, a[sm], false, b[sn], (short)0, acc[sm][sn], false, false);
    }
    __syncthreads();
  }

  // ---- Epilogue: VGPR r -> row r (+8 for upper lane half), lane lm -> col ----
#pragma unroll
  for (int sm = 0; sm < 2; ++sm) {
#pragma unroll
    for (int sn = 0; sn < 4; ++sn) {
      const int col = n0 + waveN * 64 + sn * 16 + lm;
#pragma unroll
      for (int r = 0; r < 8; ++r) {
        const int row = m0 + waveM * 32 + sm * 16 + hi * 8 + r;
        const float v = acc[sm][sn][r];
        if (FUSE) {
          const float pre = v + bias[col];
          C[(size_t)row * ldc + col] = pre;                    // pre_relu
          C2[(size_t)row * ldc + col] = pre > 0.f ? pre : 0.f; // relu out
        } else {
          C[(size_t)row * ldc + col] = v;
        }
      }
    }
  }
}

// ---------------------------------------------------------------------------
extern "C" void kernel_launch(void* const* d_in, const int* in_sizes, int n_in,
                              void* d_out, int out_size, void* d_ws, size_t ws_size,
                              hipStream_t stream) {
  const float* features = (const float*)d_in[0]; // [B, I, F]
  const float* A        = (const float*)d_in[1]; // [I, F, K]
  const float* Bm       = (const float*)d_in[2]; // [I, K, H]
  const float* b_final  = (const float*)d_in[3]; // [I, F]

  float* out = (float*)d_out;
  const size_t S_BIK = (size_t)B_SZ * N_INST * KDIM;  // == B*I*F (F==K)
  const size_t S_BIH = (size_t)B_SZ * N_INST * N_HID;
  float* o_out = out;            // [B, I, F]
  float* o_h0  = o_out + S_BIK;  // [B, I, K]
  float* o_h1  = o_h0 + S_BIK;   // [B, I, K]
  float* o_hid = o_h1 + S_BIK;   // [B, I, H]
  float* o_pre = o_hid + S_BIH;  // [B, I, F]
  float* o_nA  = o_pre + S_BIK;  // [I, F, K]

  const dim3 blk(256);

  // 0) normed_A
  norm_cols_kernel<<<dim3(KDIM / 256, N_INST), blk, 0, stream>>>(A, o_nA);

  // 1) h_0 = features @ nA            [2048 x 1024] = [2048 x 1024][1024 x 1024]
  gemm_f32_wmma_kernel<false, false>
      <<<dim3(KDIM / BN, B_SZ / BM, N_INST), blk, 0, stream>>>(
          features, (long long)N_FEAT, N_INST * N_FEAT,
          o_nA, (long long)N_FEAT * KDIM, KDIM,
          o_h0, (long long)KDIM, N_INST * KDIM,
          nullptr, nullptr, N_FEAT);

  // 2) hidden = h_0 @ B               [2048 x 256] = [2048 x 1024][1024 x 256]
  gemm_f32_wmma_kernel<false, false>
      <<<dim3(N_HID / BN, B_SZ / BM, N_INST), blk, 0, stream>>>(
          o_h0, (long long)KDIM, N_INST * KDIM,
          Bm, (long long)KDIM * N_HID, N_HID,
          o_hid, (long long)N_HID, N_INST * N_HID,
          nullptr, nullptr, KDIM);

  // 3) h_1 = hidden @ B^T             [2048 x 1024] = [2048 x 256][256 x 1024]
  gemm_f32_wmma_kernel<true, false>
      <<<dim3(KDIM / BN, B_SZ / BM, N_INST), blk, 0, stream>>>(
          o_hid, (long long)N_HID, N_INST * N_HID,
          Bm, (long long)KDIM * N_HID, N_HID,
          o_h1, (long long)KDIM, N_INST * KDIM,
          nullptr, nullptr, N_HID);

  // 4) pre_relu = h_1 @ nA^T + bias; out = relu(pre_relu)
  gemm_f32_wmma_kernel<true, true>
      <<<dim3(N_FEAT / BN, B_SZ / BM, N_INST), blk, 0, stream>>>(
          o_h1, (long long)KDIM, N_INST * KDIM,
          o_nA, (long long)N_FEAT * KDIM, KDIM,
          o_pre, (long long)N_FEAT, N_INST * N_FEAT,
          b_final, o_out, KDIM);
}